// DecoupledTopKDistillationLoss_60086592471159
// MI455X (gfx1250) — compile-verified
//
#include <hip/hip_runtime.h>
#include <hip/hip_bf16.h>
#include <float.h>
#include <stdint.h>

#define V_SIZE 32000
#define V4     (V_SIZE / 4)
#define TOPK   32
#define BS     512
#define NWAVE  (BS / 32)

#if defined(__has_builtin)
#if __has_builtin(__builtin_amdgcn_tensor_load_to_lds) && __has_builtin(__builtin_amdgcn_s_wait_tensorcnt)
#define USE_TDM 1
#endif
#endif

typedef unsigned int v4u __attribute__((ext_vector_type(4)));
typedef int          v8i __attribute__((ext_vector_type(8)));
typedef int          v4i __attribute__((ext_vector_type(4)));

// Order-preserving fp32 -> uint32 transform (larger float => larger uint).
__device__ __forceinline__ unsigned mono(float x) {
  unsigned u = __float_as_uint(x);
  return (u & 0x80000000u) ? ~u : (u | 0x80000000u);
}

__device__ __forceinline__ float wave_sum(float v) {
#pragma unroll
  for (int o = 16; o > 0; o >>= 1) v += __shfl_xor(v, o, 32);
  return v;
}
__device__ __forceinline__ float wave_max(float v) {
#pragma unroll
  for (int o = 16; o > 0; o >>= 1) v = fmaxf(v, __shfl_xor(v, o, 32));
  return v;
}

__device__ __forceinline__ void online_update(float& m, float& s, float x) {
  float M = fmaxf(m, x);
  s = s * expf(m - M) + expf(x - M);
  m = M;
}

// Block-wide merge of per-thread online-softmax partials (m = running max,
// s = sum of exp(x - m)). Result written to *outM / *outZ (shared scalars).
__device__ void softmax_block_reduce(float m, float s, float* red_m, float* red_s,
                                     float* outM, float* outZ, int tid) {
#pragma unroll
  for (int o = 16; o > 0; o >>= 1) {
    float mo = __shfl_xor(m, o, 32);
    float so = __shfl_xor(s, o, 32);
    float M  = fmaxf(m, mo);
    s = s * expf(m - M) + so * expf(mo - M);
    m = M;
  }
  if ((tid & 31) == 0) { red_m[tid >> 5] = m; red_s[tid >> 5] = s; }
  __syncthreads();
  if (tid < 32) {
    float m2 = (tid < NWAVE) ? red_m[tid] : -FLT_MAX;
    float s2 = (tid < NWAVE) ? red_s[tid] : 0.0f;
#pragma unroll
    for (int o = 16; o > 0; o >>= 1) {
      float mo = __shfl_xor(m2, o, 32);
      float so = __shfl_xor(s2, o, 32);
      float M  = fmaxf(m2, mo);
      s2 = s2 * expf(m2 - M) + so * expf(mo - M);
      m2 = M;
    }
    if (tid == 0) { *outM = m2; *outZ = s2; }
  }
  __syncthreads();
}

__global__ __launch_bounds__(BS) void tdkl_row_kernel(const float* __restrict__ sl_all,
                                                      const float* __restrict__ tl_all,
                                                      float* __restrict__ ws_bce,
                                                      float* __restrict__ ws_pt,
                                                      float* __restrict__ ws_kl) {
  __shared__ __align__(16) float s_row[V_SIZE];   // teacher row staged via TDM (128 KB)
  __shared__ float    red_m[NWAVE], red_s[NWAVE];
  __shared__ int      hist[256];
  __shared__ unsigned s_prefix;
  __shared__ int      s_need;
  __shared__ float    s_Ms, s_Zs, s_Mt, s_Zt;
  __shared__ int      s_idx[TOPK];
  __shared__ float    s_tv[TOPK];
  __shared__ int      s_cnt_gt, s_cnt_eq;

  const int tid = threadIdx.x;
  const int row = blockIdx.x;
  const float* __restrict__ sl = sl_all + (size_t)row * V_SIZE;
  const float* __restrict__ tl = tl_all + (size_t)row * V_SIZE;

  // ---- Kick off async TDM copy: teacher row (global) -> LDS.  Issued by
  // wave 0 only (TDM is a per-wave DMA op, EXEC-independent).
#ifdef USE_TDM
  if (tid < 32) {
    unsigned long long ga = (unsigned long long)(uintptr_t)tl;
    unsigned lds_off = (unsigned)(uintptr_t)(&s_row[0]);  // low 32 bits = LDS byte offset

    v4u g0;                                        // D# group 0 (ISA 8.3)
    g0.x = 1u;                                     // count = 1 (valid descriptor)
    g0.y = lds_off;                                // lds_addr
    g0.z = (unsigned)(ga & 0xFFFFFFFFull);         // global_addr[31:0]
    g0.w = ((unsigned)(ga >> 32) & 0x01FFFFFFu)    // global_addr[56:32]
           | 0x80000000u;                          // type = 2 ("image")

    v8i g1;                                        // D# group 1 (ISA 8.4)
    g1[0] = 0x00020000;                            // data_size = 2 -> 4 bytes; no wg_mask
    g1[1] = (int)((V_SIZE & 0xFFFF) << 16);        // tensor_dim0[15:0]
    g1[2] = (int)(((V_SIZE >> 16) & 0xFFFF) | (1 << 16)); // dim0 hi | tensor_dim1 = 1
    g1[3] = (int)((V_SIZE & 0xFFFF) << 16);        // tile_dim0 = 32000
    g1[4] = 1;                                     // tile_dim1 = 1, tile_dim2 = 0
    g1[5] = V_SIZE;                                // tensor_dim0_stride[31:0]
    g1[6] = (int)((V_SIZE & 0xFFFF) << 16);        // stride0 hi=0 | tensor_dim1_stride lo
    g1[7] = 0;                                     // tensor_dim1_stride hi

    v4i g2; g2[0] = 1; g2[1] = 1; g2[2] = 0; g2[3] = 0;   // dim2=1, dim3=1
    v4i g3; g3[0] = 0; g3[1] = 0; g3[2] = 0; g3[3] = 0;

#if defined(__clang_major__) && (__clang_major__ >= 23)
    v8i g4; g4[0]=0; g4[1]=0; g4[2]=0; g4[3]=0; g4[4]=0; g4[5]=0; g4[6]=0; g4[7]=0;
    __builtin_amdgcn_tensor_load_to_lds(g0, g1, g2, g3, g4, 0);
#else
    __builtin_amdgcn_tensor_load_to_lds(g0, g1, g2, g3, 0);
#endif
  }
#endif

  // ---- Student online softmax pass: b128 stream from HBM, overlapped with TDM.
  float m = -FLT_MAX, s = 0.0f;
  {
    const float4* __restrict__ sl4 = (const float4*)sl;
    for (int i = tid; i < V4; i += BS) {
      float4 x = sl4[i];
      online_update(m, s, x.x);
      online_update(m, s, x.y);
      online_update(m, s, x.z);
      online_update(m, s, x.w);
    }
  }
  softmax_block_reduce(m, s, red_m, red_s, &s_Ms, &s_Zs, tid);

  if (tid < 256) hist[tid] = 0;
  if (tid == 0) { s_prefix = 0u; s_need = TOPK; }

  // ---- Wait for the TDM transfer, then publish LDS to the workgroup.
#ifdef USE_TDM
  if (tid < 32) __builtin_amdgcn_s_wait_tensorcnt(0);
#else
  for (int i = tid; i < V_SIZE; i += BS) s_row[i] = tl[i];
#endif
  __syncthreads();

  // ---- Fused pass: teacher online softmax stats + level-0 radix histogram
  //      (top byte of the monotone uint encoding), b128 reads from LDS.
  m = -FLT_MAX; s = 0.0f;
  {
    const float4* row4 = (const float4*)s_row;
    for (int i = tid; i < V4; i += BS) {
      float4 x = row4[i];
      online_update(m, s, x.x);
      online_update(m, s, x.y);
      online_update(m, s, x.z);
      online_update(m, s, x.w);
      atomicAdd(&hist[mono(x.x) >> 24], 1);
      atomicAdd(&hist[mono(x.y) >> 24], 1);
      atomicAdd(&hist[mono(x.z) >> 24], 1);
      atomicAdd(&hist[mono(x.w) >> 24], 1);
    }
  }
  softmax_block_reduce(m, s, red_m, red_s, &s_Mt, &s_Zt, tid);
  // (barriers inside the reduce also make the level-0 histogram visible)

  if (tid == 0) {
    int need = TOPK;
    int b = 255;
    for (; b > 0; --b) {
      int c = hist[b];
      if (c >= need) break;
      need -= c;
    }
    s_prefix = (unsigned)b;
    s_need = need;
  }
  __syncthreads();

  // ---- Radix-select levels 1..3: refine to the exact 32nd-largest value.
  for (int level = 1; level < 4; ++level) {
    if (tid < 256) hist[tid] = 0;
    __syncthreads();
    unsigned pref = s_prefix;
    int sh = 24 - 8 * level;           // sh+8 <= 24, no UB
    const float4* row4 = (const float4*)s_row;
    for (int i = tid; i < V4; i += BS) {
      float4 x = row4[i];
      unsigned u0 = mono(x.x), u1 = mono(x.y), u2 = mono(x.z), u3 = mono(x.w);
      if ((u0 >> (sh + 8)) == pref) atomicAdd(&hist[(u0 >> sh) & 255], 1);
      if ((u1 >> (sh + 8)) == pref) atomicAdd(&hist[(u1 >> sh) & 255], 1);
      if ((u2 >> (sh + 8)) == pref) atomicAdd(&hist[(u2 >> sh) & 255], 1);
      if ((u3 >> (sh + 8)) == pref) atomicAdd(&hist[(u3 >> sh) & 255], 1);
    }
    __syncthreads();
    if (tid == 0) {
      int need = s_need;
      int b = 255;
      for (; b > 0; --b) {
        int c = hist[b];
        if (c >= need) break;
        need -= c;
      }
      s_prefix = (s_prefix << 8) | (unsigned)b;
      s_need = need;
    }
    __syncthreads();
  }

  // ---- Collect: all elements strictly above threshold + 'needEq' ties.
  if (tid == 0) { s_cnt_gt = 0; s_cnt_eq = 0; }
  if (tid < TOPK) { s_idx[tid] = 0; s_tv[tid] = -FLT_MAX; }
  __syncthreads();
  {
    unsigned uT = s_prefix;      // uint encoding of the 32nd-largest value
    int needEq  = s_need;        // how many == uT to take
    int base    = TOPK - needEq;
    const float4* row4 = (const float4*)s_row;
    for (int i = tid; i < V4; i += BS) {
      float4 x = row4[i];
      float v[4] = {x.x, x.y, x.z, x.w};
#pragma unroll
      for (int j = 0; j < 4; ++j) {
        unsigned u = mono(v[j]);
        if (u > uT) {
          int p = atomicAdd(&s_cnt_gt, 1);
          if (p < TOPK) { s_idx[p] = 4 * i + j; s_tv[p] = v[j]; }
        } else if (u == uT) {
          int p = atomicAdd(&s_cnt_eq, 1);
          if (p < needEq) { int q = base + p; s_idx[q] = 4 * i + j; s_tv[q] = v[j]; }
        }
      }
    }
  }
  __syncthreads();

  // ---- Per-row loss terms: one wave, one lane per top-k slot.
  if (tid < 32) {
    int   k   = tid;
    float tvk = s_tv[k];
    int   ik  = s_idx[k];
    ik = (ik < 0) ? 0 : ((ik >= V_SIZE) ? V_SIZE - 1 : ik);
    float svk = sl[ik];                  // gather student logit (L2-hot)

    // p_t, p_s on the top-k support
    float pt = wave_sum(expf(tvk - s_Mt)) / s_Zt;
    float ps = wave_sum(expf(svk - s_Ms)) / s_Zs;

    // truncated KL at temperature 2 over the 32 shared slots
    float a = tvk * 0.5f;
    float b = svk * 0.5f;
    float ma = wave_max(a);
    float mb = wave_max(b);
    float sa = wave_sum(expf(a - ma));
    float sb = wave_sum(expf(b - mb));
    float logp = a - ma - logf(sa);
    float logq = b - mb - logf(sb);
    float kl = wave_sum(expf(logp) * (logp - logq));

    if (k == 0) {
      float log_ps   = fmaxf(logf(ps),     -100.0f);
      float log_1mps = fmaxf(log1pf(-ps),  -100.0f);
      float bce = -(pt * log_ps + (1.0f - pt) * log_1mps);
      ws_bce[row] = bce;
      ws_pt[row]  = pt;
      ws_kl[row]  = kl;
    }
  }
}

// Deterministic final reduction + scalar combine:
// out = mean(bce) + mean(pt) * T^2 * (sum(kl)/N), T = 2.
__global__ __launch_bounds__(256) void tdkl_finalize(const float* __restrict__ ws, int N,
                                                     float* __restrict__ out) {
  __shared__ float rA[256], rB[256], rC[256];
  int t = threadIdx.x;
  float a = 0.0f, b = 0.0f, c = 0.0f;
  for (int i = t; i < N; i += 256) {
    a += ws[i];
    b += ws[N + i];
    c += ws[2 * N + i];
  }
  rA[t] = a; rB[t] = b; rC[t] = c;
  __syncthreads();
  for (int off = 128; off > 0; off >>= 1) {
    if (t < off) { rA[t] += rA[t + off]; rB[t] += rB[t + off]; rC[t] += rC[t + off]; }
    __syncthreads();
  }
  if (t == 0) {
    float n = (float)N;
    out[0] = rA[0] / n + (rB[0] / n) * 4.0f * (rC[0] / n);
  }
}

extern "C" void kernel_launch(void* const* d_in, const int* in_sizes, int n_in,
                              void* d_out, int out_size, void* d_ws, size_t ws_size,
                              hipStream_t stream) {
  const float* sl = (const float*)d_in[0];
  const float* tl = (const float*)d_in[1];
  float* out = (float*)d_out;
  float* ws  = (float*)d_ws;

  const int N = in_sizes[0] / V_SIZE;   // 4096 rows

  float* ws_bce = ws;
  float* ws_pt  = ws + N;
  float* ws_kl  = ws + 2 * N;

  tdkl_row_kernel<<<N, BS, 0, stream>>>(sl, tl, ws_bce, ws_pt, ws_kl);
  tdkl_finalize<<<1, 256, 0, stream>>>(ws, N, out);
}